// StiefelHouseholderCanonical_57818849739330
// MI455X (gfx1250) — compile-verified
//
#include <hip/hip_runtime.h>

// ---------------------------------------------------------------------------
// StiefelHouseholderCanonical via compact-WY:
//   G = P^T P            (WMMA f32 16x16x4)
//   M = strict_upper(G) + diag((1+G_ii)/2) ;  T = M^{-1} (upper-tri solve)
//   X_top    = I - T     (written by the inversion kernel)
//   X_bottom = -P @ T    (WMMA f32 16x16x4; T recovered as I - X_top)
// Turns the reference's 128-pass sequential scan (~4.3 GB traffic) into two
// small fp32 GEMMs + a 128x128 triangular inversion (~32 MB traffic total).
// ---------------------------------------------------------------------------

typedef __attribute__((ext_vector_type(2))) float v2f;
typedef __attribute__((ext_vector_type(8))) float v8f;

#define BATCH 32
#define NDIM  1024
#define RANK  128
#define NMR   896   // NDIM - RANK

static __device__ __forceinline__ v8f wmma4(v2f a, v2f b, v8f c) {
  // V_WMMA_F32_16X16X4_F32 : D = A(16x4) * B(4x16) + C(16x16), all fp32.
  // 8-arg form: (neg_a, A, neg_b, B, c_mod, C, reuse_a, reuse_b).
  return __builtin_amdgcn_wmma_f32_16x16x4_f32(false, a, false, b,
                                               (short)0, c, false, false);
}

// ---------------------------------------------------------------------------
// Phase 1: G[b] = P[b]^T @ P[b]  (RANK x RANK). One wave per 32x32 block of G.
// A[m][k] = P[k][bi+m], B[k][n] = P[k][bj+n].
// ISA 16x4 fp32 A layout: lane L holds M = L%16, K = 2*(L/16) + {vgpr0,vgpr1}.
// ---------------------------------------------------------------------------
__global__ void __launch_bounds__(32)
k_gram(const float* __restrict__ P, float* __restrict__ G) {
  const int batch = blockIdx.y;
  const int bi = (blockIdx.x >> 2) << 5;   // 0,32,64,96
  const int bj = (blockIdx.x & 3) << 5;
  const int lane = threadIdx.x;
  const int m  = lane & 15;
  const int kk = (lane >> 4) << 1;         // 0 or 2
  const float* Pb = P + (size_t)batch * NMR * RANK;

  v8f c00 = {}, c01 = {}, c10 = {}, c11 = {};
#pragma unroll 2
  for (int k0 = 0; k0 < NMR; k0 += 4) {
    const float* r0 = Pb + (size_t)(k0 + kk) * RANK;   // K = k0+kk   row of P
    const float* r1 = r0 + RANK;                       // K = k0+kk+1
    v2f a0 = { r0[bi + m],      r1[bi + m]      };
    v2f a1 = { r0[bi + 16 + m], r1[bi + 16 + m] };
    v2f b0 = { r0[bj + m],      r1[bj + m]      };
    v2f b1 = { r0[bj + 16 + m], r1[bj + 16 + m] };
    c00 = wmma4(a0, b0, c00);
    c01 = wmma4(a0, b1, c01);
    c10 = wmma4(a1, b0, c10);
    c11 = wmma4(a1, b1, c11);
  }

  // C/D layout: VGPR v, lanes 0-15 -> (M=v, N=lane); lanes 16-31 -> (M=v+8).
  float* Gb = G + (size_t)batch * RANK * RANK;
  const int mb = (lane >> 4) << 3;
#pragma unroll
  for (int v = 0; v < 8; ++v) {
    Gb[(bi + mb + v) * RANK + bj + m]           = c00[v];
    Gb[(bi + mb + v) * RANK + bj + 16 + m]      = c01[v];
    Gb[(bi + 16 + mb + v) * RANK + bj + m]      = c10[v];
    Gb[(bi + 16 + mb + v) * RANK + bj + 16 + m] = c11[v];
  }
}

// ---------------------------------------------------------------------------
// Phase 2: per batch, T = M^{-1} with M = strict_upper(G) + diag((1+G_ii)/2)
// (T^{-1} = diag(1/beta) + strict_upper(V^T V), beta_i = 2/(1+||p_i||^2)).
// Thread t back-substitutes column t; X_top = I - T goes to d_out rows 0..127.
// 129-float LDS row padding keeps both access patterns bank-conflict free.
// ---------------------------------------------------------------------------
__global__ void __launch_bounds__(128)
k_invtop(const float* __restrict__ G, float* __restrict__ out) {
  __shared__ float sM[RANK][RANK + 1];     // upper-triangular part of M
  __shared__ float sT[RANK][RANK + 1];     // sT[col][row]
  __shared__ float sRd[RANK];              // 1 / M[i][i] = beta_i
  const int b = blockIdx.x;
  const int t = threadIdx.x;               // 0..127 : column of T
  const float* Gb = G + (size_t)b * RANK * RANK;

  for (int i = 0; i < RANK; ++i) {
    float g = Gb[i * RANK + t];
    sM[i][t] = (i < t) ? g : 0.0f;
    if (i == t) sRd[t] = 2.0f / (1.0f + g);    // M_tt = (1+G_tt)/2
  }
  __syncthreads();

  float* tc = sT[t];
  for (int j = 0; j < RANK; ++j) tc[j] = 0.0f;
  tc[t] = sRd[t];                              // T[t][t] = beta_t
  for (int i = RANK - 2; i >= 0; --i) {
    if (i < t) {
      float s = 0.0f;
      for (int j = i + 1; j <= t; ++j) s += sM[i][j] * tc[j];
      tc[i] = -s * sRd[i];
    }
  }
  __syncthreads();

  float* Ob = out + (size_t)b * NDIM * RANK;   // rows 0..127 of X
  for (int i = 0; i < RANK; ++i) {
    float v = sT[t][i];                        // T[i][t]
    Ob[i * RANK + t] = ((i == t) ? 1.0f : 0.0f) - v;
  }
}

// ---------------------------------------------------------------------------
// Phase 3: X_bottom = -P @ T, with T recovered as I - X_top from d_out.
// One wave per 32x32 output block; K = RANK = 128 (32 WMMA steps x 4 tiles).
// ---------------------------------------------------------------------------
__global__ void __launch_bounds__(32)
k_xbottom(const float* __restrict__ P, const float* __restrict__ xtop,
          float* __restrict__ out) {
  const int batch = blockIdx.z;
  const int bi = blockIdx.y << 5;          // 0..864 (28 blocks over 896 rows)
  const int bj = blockIdx.x << 5;          // 0,32,64,96
  const int lane = threadIdx.x;
  const int m  = lane & 15;
  const int kk = (lane >> 4) << 1;
  const float* Pb = P + (size_t)batch * NMR * RANK;
  const float* Xt = xtop + (size_t)batch * NDIM * RANK;  // rows 0..127 = I - T

  v8f c00 = {}, c01 = {}, c10 = {}, c11 = {};
#pragma unroll 2
  for (int k0 = 0; k0 < RANK; k0 += 4) {
    const int kr = k0 + kk;
    // A[m][k] = P[bi+m][k] : two consecutive fp32 along K -> one v2f load.
    v2f a0 = *(const v2f*)(Pb + (size_t)(bi + m) * RANK + kr);
    v2f a1 = *(const v2f*)(Pb + (size_t)(bi + 16 + m) * RANK + kr);
    // B[k][n] = T[k][bj+n] = delta(k, bj+n) - Xtop[k][bj+n]
    const float* x0 = Xt + (size_t)kr * RANK;
    const float* x1 = x0 + RANK;
    v2f b0, b1;
    b0.x = ((kr     == bj + m)      ? 1.0f : 0.0f) - x0[bj + m];
    b0.y = ((kr + 1 == bj + m)      ? 1.0f : 0.0f) - x1[bj + m];
    b1.x = ((kr     == bj + 16 + m) ? 1.0f : 0.0f) - x0[bj + 16 + m];
    b1.y = ((kr + 1 == bj + 16 + m) ? 1.0f : 0.0f) - x1[bj + 16 + m];
    c00 = wmma4(a0, b0, c00);
    c01 = wmma4(a0, b1, c01);
    c10 = wmma4(a1, b0, c10);
    c11 = wmma4(a1, b1, c11);
  }

  // X rows 128.. : store -(P@T). (f32 WMMA has no A/B NEG bits, so negate here.)
  float* Ob = out + (size_t)batch * NDIM * RANK + (size_t)RANK * RANK;
  const int mb = (lane >> 4) << 3;
#pragma unroll
  for (int v = 0; v < 8; ++v) {
    Ob[(bi + mb + v) * RANK + bj + m]           = -c00[v];
    Ob[(bi + mb + v) * RANK + bj + 16 + m]      = -c01[v];
    Ob[(bi + 16 + mb + v) * RANK + bj + m]      = -c10[v];
    Ob[(bi + 16 + mb + v) * RANK + bj + 16 + m] = -c11[v];
  }
}

// ---------------------------------------------------------------------------
extern "C" void kernel_launch(void* const* d_in, const int* in_sizes, int n_in,
                              void* d_out, int out_size, void* d_ws, size_t ws_size,
                              hipStream_t stream) {
  (void)in_sizes; (void)n_in; (void)out_size; (void)ws_size;
  const float* P = (const float*)d_in[0];        // (32, 896, 128) fp32
  float* out = (float*)d_out;                    // (32, 1024, 128) fp32
  float* G   = (float*)d_ws;                     // 32*128*128 fp32 = 2 MB scratch

  k_gram   <<<dim3(16, BATCH),   dim3(32),  0, stream>>>(P, G);
  k_invtop <<<dim3(BATCH),       dim3(128), 0, stream>>>(G, out);
  k_xbottom<<<dim3(4, 28, BATCH), dim3(32), 0, stream>>>(P, out, out);
}